// CrossAttention_80427557585105
// MI455X (gfx1250) — compile-verified
//
#include <hip/hip_runtime.h>
#include <math.h>

typedef __attribute__((ext_vector_type(2))) float v2f;
typedef __attribute__((ext_vector_type(8))) float v8f;

#define HEADS 16
#define DHEAD 64
#define INNER 1024
#define NQ 2048      // query length
#define NM 1024      // context length (= K for fw modulation)
#define QD 1024
#define CDIM 768
#define SCALE_F 0.125f   // 64^-0.5
#define STEP_T 0.6f

#define ROWS 64          // rows per attention workgroup (4 WMMA row-tiles)
#define RT 4             // row-tiles per wave sharing one B fragment
#define ATT_THREADS 512  // 16 waves

__device__ __forceinline__ v8f wmma4(v2f a, v2f b, v8f c) {
  // V_WMMA_F32_16X16X4_F32 : D = A(16x4) * B(4x16) + C
  return __builtin_amdgcn_wmma_f32_16x16x4_f32(false, a, false, b, (short)0, c,
                                               false, false);
}

// ---------------------------------------------------------------------------
// Generic GEMM: Out = In(Mr x K) * W^T  (W is INNER x K row-major).
// One wave -> 64x16 output (4 row-tiles sharing each B fragment).
// 8 waves (256 thr) per block -> 512 rows per block.
// REMAP scatters the output into per-head (b*H+h, seq, 64) layout.
// ---------------------------------------------------------------------------
template <bool REMAP, bool BIAS>
__global__ __launch_bounds__(256) void proj_gemm(
    const float* __restrict__ In, const float* __restrict__ W,
    const float* __restrict__ bias, float* __restrict__ Out,
    int Mr, int K, int seq) {
  const int wave = threadIdx.x >> 5;
  const int lane = threadIdx.x & 31;
  const int l16  = lane & 15;
  const int hi   = lane >> 4;            // 0 or 1
  const int col0 = blockIdx.x * 16;
  const int row0 = (blockIdx.y * 8 + wave) * (RT * 16);
  if (row0 >= Mr) return;                // wave-uniform

  const float* Brow = W + (size_t)(col0 + l16) * K + 2 * hi; // B^T access
  v8f acc[RT] = {};
#pragma unroll 4
  for (int k0 = 0; k0 < K; k0 += 4) {
    v2f b = *(const v2f*)(Brow + k0);
#pragma unroll
    for (int rt = 0; rt < RT; ++rt) {
      v2f a = *(const v2f*)(In + (size_t)(row0 + rt * 16 + l16) * K + k0 + 2 * hi);
      acc[rt] = wmma4(a, b, acc[rt]);
    }
  }
#pragma unroll
  for (int rt = 0; rt < RT; ++rt) {
#pragma unroll
    for (int r = 0; r < 8; ++r) {
      int row = row0 + rt * 16 + r + 8 * hi;
      int col = col0 + l16;
      float val = acc[rt][r];
      if (BIAS) val += bias[col];
      if (REMAP) {
        int b = row / seq, n = row % seq;
        int h = col >> 6, d = col & 63;
        Out[(((size_t)(b * HEADS + h) * seq) + n) * DHEAD + d] = val;
      } else {
        Out[(size_t)row * INNER + col] = val;
      }
    }
  }
}

// ---------------------------------------------------------------------------
// Compute one 64 x 1024 sim tile (scaled) into LDS.
// sim[n][m] = SCALE * sum_d q[n][d] * k[m][d]   (B = k^T, trans access)
// 16 waves x 4 column-tiles each cover all 64 m-tiles; each B fragment
// feeds 4 row-tiles.
// ---------------------------------------------------------------------------
__device__ __forceinline__ void sim_tile64(const float* __restrict__ qg,
                                           const float* __restrict__ kg,
                                           float* S, int bh, int n0,
                                           int wave, int l16, int hi) {
  const float* qp = qg + ((size_t)bh * NQ + n0) * DHEAD;
  const float* kp = kg + (size_t)bh * NM * DHEAD;
  for (int ct = 0; ct < 4; ++ct) {
    const int m0 = (wave * 4 + ct) * 16;
    const float* Brow = kp + (size_t)(m0 + l16) * DHEAD + 2 * hi;
    v8f acc[RT] = {};
#pragma unroll
    for (int k0 = 0; k0 < DHEAD; k0 += 4) {
      v2f b = *(const v2f*)(Brow + k0);
#pragma unroll
      for (int rt = 0; rt < RT; ++rt) {
        v2f a = *(const v2f*)(qp + (size_t)(rt * 16 + l16) * DHEAD + k0 + 2 * hi);
        acc[rt] = wmma4(a, b, acc[rt]);
      }
    }
#pragma unroll
    for (int rt = 0; rt < RT; ++rt)
#pragma unroll
      for (int r = 0; r < 8; ++r)
        S[(rt * 16 + r + 8 * hi) * NM + m0 + l16] = acc[rt][r] * SCALE_F;
  }
}

// fwraw[n][d] = sum_k S[n][k] * fam[d][k]  (B = fam^T; one B fragment -> 4 row-tiles)
__device__ __forceinline__ void fw_tile64(const float* S,
                                          const float* __restrict__ fam,
                                          int d0, int l16, int hi, v8f out[RT]) {
  const float* Brow = fam + (size_t)(d0 + l16) * NM + 2 * hi;
#pragma unroll
  for (int rt = 0; rt < RT; ++rt) out[rt] = {};
#pragma unroll 2
  for (int k0 = 0; k0 < NM; k0 += 4) {
    v2f b = *(const v2f*)(Brow + k0);
#pragma unroll
    for (int rt = 0; rt < RT; ++rt) {
      v2f a = *(const v2f*)(S + (size_t)(rt * 16 + l16) * NM + k0 + 2 * hi);
      out[rt] = wmma4(a, b, out[rt]);
    }
  }
}

// ---------------------------------------------------------------------------
// Pass A: per (bh, 64-row tile) compute rowmax_d( min(|fw|,1) ).
// ---------------------------------------------------------------------------
__global__ __launch_bounds__(ATT_THREADS) void fwmax_kernel(
    const float* __restrict__ q, const float* __restrict__ k,
    const float* __restrict__ fam, float* __restrict__ fwmax) {
  __shared__ float S[ROWS * NM];       // 256 KB
  __shared__ int rowmax[ROWS];
  const int wave = threadIdx.x >> 5;
  const int lane = threadIdx.x & 31;
  const int l16 = lane & 15, hi = lane >> 4;
  const int bh = blockIdx.y, n0 = blockIdx.x * ROWS;

  if (threadIdx.x < ROWS) rowmax[threadIdx.x] = 0;
  sim_tile64(q, k, S, bh, n0, wave, l16, hi);
  __syncthreads();

  float lmax[RT][8];
#pragma unroll
  for (int rt = 0; rt < RT; ++rt)
#pragma unroll
    for (int r = 0; r < 8; ++r) lmax[rt][r] = 0.0f;

  for (int ct = 0; ct < 4; ++ct) {
    v8f f[RT];
    fw_tile64(S, fam, (wave * 4 + ct) * 16, l16, hi, f);
#pragma unroll
    for (int rt = 0; rt < RT; ++rt)
#pragma unroll
      for (int r = 0; r < 8; ++r)
        lmax[rt][r] = fmaxf(lmax[rt][r], fminf(fabsf(f[rt][r]), 1.0f));
  }
#pragma unroll
  for (int rt = 0; rt < RT; ++rt)
#pragma unroll
    for (int r = 0; r < 8; ++r)
      atomicMax(&rowmax[rt * 16 + r + 8 * hi], __float_as_int(lmax[rt][r]));
  __syncthreads();
  if (threadIdx.x < ROWS)
    fwmax[(size_t)bh * NQ + n0 + threadIdx.x] = __int_as_float(rowmax[threadIdx.x]);
}

// ---------------------------------------------------------------------------
// Pass B: recompute sim + fw, threshold-adjust, softmax, attn @ V.
// ---------------------------------------------------------------------------
__global__ __launch_bounds__(ATT_THREADS) void attn_kernel(
    const float* __restrict__ q, const float* __restrict__ k,
    const float* __restrict__ v, const float* __restrict__ fam,
    const float* __restrict__ fwmax, float* __restrict__ outh) {
  __shared__ float S[ROWS * NM];               // 256 KB
  __shared__ unsigned int keep[ROWS][NM / 32]; // 8 KB threshold bitmask
  __shared__ float red[ROWS][9];
  __shared__ float fwm[ROWS];
  const int wave = threadIdx.x >> 5;
  const int lane = threadIdx.x & 31;
  const int l16 = lane & 15, hi = lane >> 4;
  const int bh = blockIdx.y, n0 = blockIdx.x * ROWS;

  for (int i = threadIdx.x; i < ROWS * (NM / 32); i += ATT_THREADS)
    ((unsigned int*)keep)[i] = 0u;
  if (threadIdx.x < ROWS)
    fwm[threadIdx.x] = fwmax[(size_t)bh * NQ + n0 + threadIdx.x];
  sim_tile64(q, k, S, bh, n0, wave, l16, hi);
  __syncthreads();

  // fw tiles; record "keep" bits (writes go to keep[], S stays read-only here)
  for (int ct = 0; ct < 4; ++ct) {
    const int d0 = (wave * 4 + ct) * 16;
    v8f f[RT];
    fw_tile64(S, fam, d0, l16, hi, f);
#pragma unroll
    for (int rt = 0; rt < RT; ++rt)
#pragma unroll
      for (int r = 0; r < 8; ++r) {
        int row = rt * 16 + r + 8 * hi;
        int col = d0 + l16;
        float fn = fminf(fabsf(f[rt][r]), 1.0f) / (fwm[row] + 1e-6f);
        if (fn > STEP_T) atomicOr(&keep[row][col >> 5], 1u << (col & 31));
      }
  }
  __syncthreads();

  // sim += (fw_bin - 1) * 50  -> subtract 50 where keep bit unset
  for (int idx = threadIdx.x; idx < ROWS * NM; idx += ATT_THREADS) {
    int row = idx >> 10, col = idx & (NM - 1);
    if (!((keep[row][col >> 5] >> (col & 31)) & 1u)) S[idx] -= 50.0f;
  }
  __syncthreads();

  // softmax over 1024 cols; 8 threads per row
  const int row = threadIdx.x >> 3;
  const int sub = threadIdx.x & 7;
  float m = -1e30f;
  for (int c = sub; c < NM; c += 8) m = fmaxf(m, S[row * NM + c]);
  red[row][sub] = m;
  __syncthreads();
  if (sub == 0) {
    float mm = red[row][0];
    for (int i = 1; i < 8; ++i) mm = fmaxf(mm, red[row][i]);
    red[row][8] = mm;
  }
  __syncthreads();
  float rmax = red[row][8];
  float s = 0.0f;
  for (int c = sub; c < NM; c += 8) {
    float e = expf(S[row * NM + c] - rmax);
    S[row * NM + c] = e;
    s += e;
  }
  __syncthreads();
  red[row][sub] = s;
  __syncthreads();
  if (sub == 0) {
    float ss = 0.0f;
    for (int i = 0; i < 8; ++i) ss += red[row][i];
    red[row][8] = ss;
  }
  __syncthreads();
  float inv = 1.0f / red[row][8];
  for (int c = sub; c < NM; c += 8) S[row * NM + c] *= inv;
  __syncthreads();

  // out(64x64) = P(64x1024) @ v(1024x64); 16 waves -> 16 (rt, dt) tiles
  {
    const int rt = wave >> 2, dt = wave & 3;
    const int d0 = dt * 16;
    const float* vp = v + (size_t)bh * NM * DHEAD;
    const float* Arow = S + (size_t)(rt * 16 + l16) * NM + 2 * hi;
    v8f acc = {};
#pragma unroll 4
    for (int k0 = 0; k0 < NM; k0 += 4) {
      v2f a = *(const v2f*)(Arow + k0);
      v2f b;
      b.x = vp[(size_t)(k0 + 2 * hi) * DHEAD + d0 + l16];
      b.y = vp[(size_t)(k0 + 2 * hi + 1) * DHEAD + d0 + l16];
      acc = wmma4(a, b, acc);
    }
    const int b_ = bh >> 4, h = bh & 15;
#pragma unroll
    for (int r = 0; r < 8; ++r) {
      int n = n0 + rt * 16 + r + 8 * hi;
      outh[((size_t)(b_ * NQ + n)) * INNER + h * DHEAD + d0 + l16] = acc[r];
    }
  }
}

// ---------------------------------------------------------------------------
extern "C" void kernel_launch(void* const* d_in, const int* in_sizes, int n_in,
                              void* d_out, int out_size, void* d_ws, size_t ws_size,
                              hipStream_t stream) {
  const float* x   = (const float*)d_in[0];   // (2, 2048, 1024)
  const float* ctx = (const float*)d_in[1];   // (2, 1024, 768)
  const float* fam = (const float*)d_in[2];   // (1, 1024, 1024) -> use [0]
  const float* Wq  = (const float*)d_in[3];   // (1024, 1024)
  const float* Wk  = (const float*)d_in[4];   // (1024, 768)
  const float* Wv  = (const float*)d_in[5];   // (1024, 768)
  const float* Wo  = (const float*)d_in[6];   // (1024, 1024)
  const float* bo  = (const float*)d_in[7];   // (1024,)
  float* out = (float*)d_out;

  float* ws   = (float*)d_ws;
  float* q    = ws;                              // 32*2048*64  = 4M floats
  float* kbuf = q    + (size_t)32 * NQ * DHEAD;  // 32*1024*64  = 2M
  float* vbuf = kbuf + (size_t)32 * NM * DHEAD;  // 2M
  float* fwmx = vbuf + (size_t)32 * NM * DHEAD;  // 32*2048     = 64K
  float* outh = fwmx + (size_t)32 * NQ;          // 4096*1024   = 4M
  (void)in_sizes; (void)n_in; (void)out_size; (void)ws_size;

  dim3 blk(256);
  // Q/K/V projections (scatter into per-head layout); 512 rows per block
  proj_gemm<true, false><<<dim3(64, 8), blk, 0, stream>>>(x,   Wq, nullptr, q,    4096, QD,   NQ);
  proj_gemm<true, false><<<dim3(64, 4), blk, 0, stream>>>(ctx, Wk, nullptr, kbuf, 2048, CDIM, NM);
  proj_gemm<true, false><<<dim3(64, 4), blk, 0, stream>>>(ctx, Wv, nullptr, vbuf, 2048, CDIM, NM);
  // Pass A: row-max of focus weights (64-row tiles)
  fwmax_kernel<<<dim3(NQ / ROWS, 32), dim3(ATT_THREADS), 0, stream>>>(q, kbuf, fam, fwmx);
  // Pass B: threshold + softmax + attn @ V
  attn_kernel<<<dim3(NQ / ROWS, 32), dim3(ATT_THREADS), 0, stream>>>(q, kbuf, vbuf, fam, fwmx, outh);
  // Output projection + bias
  proj_gemm<false, true><<<dim3(64, 8), blk, 0, stream>>>(outh, Wo, bo, out, 4096, INNER, 0);
}